// Block_78864189489800
// MI455X (gfx1250) — compile-verified
//
#include <hip/hip_runtime.h>
#include <hip/hip_bf16.h>
#include <math.h>
#include <stdint.h>

typedef __attribute__((ext_vector_type(2))) float v2f;
typedef __attribute__((ext_vector_type(8))) float v8f;
typedef int v4i __attribute__((vector_size(16)));   // matches async-LDS builtin param type

#define T_DIM 384
#define C_DIM 512
#define H_DIM 8
#define D_DIM 64
#define E_DIM 16
#define LSTRIDE 68   // LDS row stride in floats: conflict-free (68%64==4) and 16B aligned

#define AS1 __attribute__((address_space(1)))
#define AS3 __attribute__((address_space(3)))

#if defined(__has_builtin)
#if __has_builtin(__builtin_amdgcn_global_load_async_to_lds_b128)
#define HAVE_ASYNC_LDS 1
#endif
#endif
#ifndef HAVE_ASYNC_LDS
#define HAVE_ASYNC_LDS 0
#endif

// 16-byte global -> LDS copy; async (ASYNCcnt) when the gfx1250 builtin exists.
__device__ __forceinline__ void copy16_to_lds(const float* gsrc, float* lds_dst) {
#if HAVE_ASYNC_LDS
  __builtin_amdgcn_global_load_async_to_lds_b128(
      (AS1 v4i*)(uintptr_t)gsrc,
      (AS3 v4i*)(unsigned int)(uintptr_t)lds_dst, 0, 0);
#else
  *(float4*)lds_dst = *(const float4*)gsrc;
#endif
}

__device__ __forceinline__ void wait_async_lds() {
#if HAVE_ASYNC_LDS
#if defined(__has_builtin) && __has_builtin(__builtin_amdgcn_s_wait_asynccnt)
  __builtin_amdgcn_s_wait_asynccnt(0);
#else
  asm volatile("s_wait_asynccnt 0x0" ::: "memory");
#endif
#endif
}

// ---------------------------------------------------------------------------
// LayerNorm: one block (256 threads) per row of 512
// ---------------------------------------------------------------------------
__global__ void ln_kernel(const float* __restrict__ x,
                          const float* __restrict__ w,
                          const float* __restrict__ b,
                          float* __restrict__ out) {
  __shared__ float red[256];
  const int row = blockIdx.x;
  const int tid = threadIdx.x;
  const float* xr = x + row * C_DIM;
  float v0 = xr[tid];
  float v1 = xr[tid + 256];

  red[tid] = v0 + v1;
  __syncthreads();
  for (int off = 128; off > 0; off >>= 1) {
    if (tid < off) red[tid] += red[tid + off];
    __syncthreads();
  }
  float mu = red[0] * (1.0f / C_DIM);
  __syncthreads();

  float d0 = v0 - mu, d1 = v1 - mu;
  red[tid] = d0 * d0 + d1 * d1;
  __syncthreads();
  for (int off = 128; off > 0; off >>= 1) {
    if (tid < off) red[tid] += red[tid + off];
    __syncthreads();
  }
  float rstd = rsqrtf(red[0] * (1.0f / C_DIM) + 1e-5f);

  out[row * C_DIM + tid]       = d0 * rstd * w[tid]       + b[tid];
  out[row * C_DIM + tid + 256] = d1 * rstd * w[tid + 256] + b[tid + 256];
}

// ---------------------------------------------------------------------------
// WMMA fp32 GEMM: out[M,N] = act(A[M,K] @ W[N,K]^T + bias[N]) (+ res[M,N])
// 256 threads = 8 waves arranged 4(M) x 2(N); each wave owns a 16x64 strip
// (1 A-fragment reused across 4 B-fragments -> 4 accumulators).
// Inner loop batches 4 k-steps: 20 clustered b64 loads, then 16 WMMAs.
// ---------------------------------------------------------------------------
template<int ACT_GELU, int HAS_RES>
__global__ __launch_bounds__(256, 1)
void gemm_kernel(const float* __restrict__ A,
                 const float* __restrict__ W,
                 const float* __restrict__ bias,
                 const float* __restrict__ res,
                 float* __restrict__ out,
                 int M, int N, int K) {
  const int lane = threadIdx.x & 31;
  const int wave = threadIdx.x >> 5;
  const int row0 = blockIdx.y * 64 + (wave >> 1) * 16;
  const int col0 = blockIdx.x * 128 + (wave & 1) * 64;
  if (row0 >= M) return;

  const int mn  = lane & 15;          // M index (A) / N index (B)
  const int khi = (lane >> 4) << 1;   // K sub-offset: lanes 0-15 -> {0,1}, 16-31 -> {2,3}

  const float* arow = A + (size_t)(row0 + mn) * K + khi;
  const float* w0   = W + (size_t)(col0 + mn) * K + khi;
  const float* w1   = w0 + (size_t)16 * K;
  const float* w2   = w0 + (size_t)32 * K;
  const float* w3   = w0 + (size_t)48 * K;

  v8f acc0 = {0.f,0.f,0.f,0.f,0.f,0.f,0.f,0.f};
  v8f acc1 = acc0, acc2 = acc0, acc3 = acc0;

  for (int k = 0; k < K; k += 16) {
    v2f af[4], bf0[4], bf1[4], bf2[4], bf3[4];
#pragma unroll
    for (int u = 0; u < 4; ++u) {
      const int kk = k + 4 * u;
      af[u]  = *(const v2f*)(arow + kk);
      bf0[u] = *(const v2f*)(w0 + kk);
      bf1[u] = *(const v2f*)(w1 + kk);
      bf2[u] = *(const v2f*)(w2 + kk);
      bf3[u] = *(const v2f*)(w3 + kk);
    }
#pragma unroll
    for (int u = 0; u < 4; ++u) {
      acc0 = __builtin_amdgcn_wmma_f32_16x16x4_f32(false, af[u], false, bf0[u], (short)0, acc0, false, false);
      acc1 = __builtin_amdgcn_wmma_f32_16x16x4_f32(false, af[u], false, bf1[u], (short)0, acc1, false, false);
      acc2 = __builtin_amdgcn_wmma_f32_16x16x4_f32(false, af[u], false, bf2[u], (short)0, acc2, false, false);
      acc3 = __builtin_amdgcn_wmma_f32_16x16x4_f32(false, af[u], false, bf3[u], (short)0, acc3, false, false);
    }
  }

  // C/D layout: VGPR r -> row (row0 + r + 8*(lane>>4)), col (tile_col0 + lane&15)
  const int rbase = row0 + ((lane >> 4) << 3);
#pragma unroll
  for (int t = 0; t < 4; ++t) {
    const v8f acc = (t == 0) ? acc0 : (t == 1) ? acc1 : (t == 2) ? acc2 : acc3;
    const int coln = col0 + t * 16 + mn;
    const float bv = bias[coln];
#pragma unroll
    for (int r = 0; r < 8; ++r) {
      float v = acc[r] + bv;
      if (ACT_GELU) v = 0.5f * v * (1.0f + erff(v * 0.70710678118f));
      size_t idx = (size_t)(rbase + r) * N + coln;
      if (HAS_RES) v += res[idx];
      out[idx] = v;
    }
  }
}

// ---------------------------------------------------------------------------
// Edge-modulated causal attention with online softmax.
// Grid: (T/16 i-tiles, H heads). Block: 256 threads.
// score[i,j] = (1/8) * sum_d q[i,d]*k[j,d]*ek[e_ij,d]  + ab[e_ij]
// y[i,d]     = sum_j softmax(score)[i,j] * v[j,d] * ev[e_ij,d]
// Thread t: i = t>>4. Score phase: j = t&15. Accum phase: d = (t&15)+16c.
// K/V tiles double-buffered via GLOBAL_LOAD_ASYNC_TO_LDS_B128: the stage of
// tile jt+1 is issued right after the barrier that publishes tile jt, so the
// ASYNCcnt DMA overlaps score/softmax/PV compute.
// ---------------------------------------------------------------------------
__global__ void attn_kernel(const float* __restrict__ qkv,
                            const int*   __restrict__ bias_mat,
                            const float* __restrict__ edge_k,
                            const float* __restrict__ edge_v,
                            const float* __restrict__ attn_bias_emb,
                            float* __restrict__ yout) {
  __shared__ float q_s [16 * LSTRIDE];
  __shared__ float k_s [2][16 * LSTRIDE];
  __shared__ float v_s [2][16 * LSTRIDE];
  __shared__ float ek_s[16 * LSTRIDE];
  __shared__ float ev_s[16 * LSTRIDE];
  __shared__ float p_s [16 * 17];
  __shared__ int   e_s [16 * 17];
  __shared__ float ab_s[E_DIM];

  const int h   = blockIdx.y;
  const int it  = blockIdx.x;
  const int tid = threadIdx.x;
  const int i   = tid >> 4;     // row within i-tile
  const int j   = tid & 15;     // col within j-tile / d-lane in accum phase
  const int i_glob = it * 16 + i;

  // one 16B chunk per thread covers a 16x64 tile
  const int ldr = tid >> 4;          // tile row 0..15
  const int ldd = (tid & 15) << 2;   // d offset 0..60
  const int lofs = ldr * LSTRIDE + ldd;

  copy16_to_lds(edge_k + ldr * C_DIM + h * D_DIM + ldd, ek_s + lofs);
  copy16_to_lds(edge_v + ldr * C_DIM + h * D_DIM + ldd, ev_s + lofs);
  copy16_to_lds(qkv + (size_t)(it * 16 + ldr) * (3 * C_DIM) + h * D_DIM + ldd,
                q_s + lofs);
  if (tid < E_DIM) ab_s[tid] = attn_bias_emb[tid * H_DIM + h];

  // prologue: stage K/V tile jt=0 into buffer 0
  {
    const float* base = qkv + (size_t)ldr * (3 * C_DIM) + h * D_DIM + ldd;
    copy16_to_lds(base + C_DIM,     k_s[0] + lofs);
    copy16_to_lds(base + 2 * C_DIM, v_s[0] + lofs);
  }

  float o0 = 0.f, o1 = 0.f, o2 = 0.f, o3 = 0.f;
  float m_i = -INFINITY, l_i = 0.f;

  for (int jt = 0; jt <= it; ++jt) {
    const int cur = jt & 1;
    const int j_glob = jt * 16 + j;
    const int e = bias_mat[i_glob * T_DIM + j_glob];
    e_s[i * 17 + j] = e;
    wait_async_lds();           // tile jt (and q/ek/ev on first pass) resident
    __syncthreads();

    // overlap: stage tile jt+1 into the other buffer while we compute on jt
    if (jt < it) {
      const float* base = qkv + (size_t)((jt + 1) * 16 + ldr) * (3 * C_DIM)
                          + h * D_DIM + ldd;
      copy16_to_lds(base + C_DIM,     k_s[cur ^ 1] + lofs);
      copy16_to_lds(base + 2 * C_DIM, v_s[cur ^ 1] + lofs);
    }

    // ---- modulated score (thread = one (i,j) pair) ----
    float s = 0.f;
    {
      const float* qp = q_s       + i * LSTRIDE;
      const float* kp = k_s[cur]  + j * LSTRIDE;
      const float* ep = ek_s      + e * LSTRIDE;
#pragma unroll 8
      for (int d = 0; d < D_DIM; ++d)
        s = fmaf(qp[d] * kp[d], ep[d], s);
    }
    s = s * 0.125f + ab_s[e];                // /sqrt(64) + attn bias
    if (j_glob > i_glob) s = -INFINITY;      // causal (diag kept)

    // ---- online softmax over the 16 j-lanes of this row ----
    float mt = s;
#pragma unroll
    for (int off = 8; off > 0; off >>= 1)
      mt = fmaxf(mt, __shfl_xor(mt, off, 32));
    const float m_new = fmaxf(m_i, mt);
    const float p = expf(s - m_new);
    float psum = p;
#pragma unroll
    for (int off = 8; off > 0; off >>= 1)
      psum += __shfl_xor(psum, off, 32);
    const float alpha = expf(m_i - m_new);
    l_i = l_i * alpha + psum;
    m_i = m_new;
    p_s[i * 17 + j] = p;
    o0 *= alpha; o1 *= alpha; o2 *= alpha; o3 *= alpha;
    __syncthreads();

    // ---- modulated PV accumulate (thread owns row i, d = j + 16c) ----
#pragma unroll
    for (int jj = 0; jj < 16; ++jj) {
      const float pj = p_s[i * 17 + jj];
      const int   ej = e_s[i * 17 + jj];
      const float* vp  = v_s[cur] + jj * LSTRIDE;
      const float* evp = ev_s     + ej * LSTRIDE;
      o0 = fmaf(pj * vp[j],      evp[j],      o0);
      o1 = fmaf(pj * vp[j + 16], evp[j + 16], o1);
      o2 = fmaf(pj * vp[j + 32], evp[j + 32], o2);
      o3 = fmaf(pj * vp[j + 48], evp[j + 48], o3);
    }
    __syncthreads();
  }

  const float inv_l = 1.0f / l_i;
  float* yr = yout + (size_t)i_glob * C_DIM + h * D_DIM;
  yr[j]      = o0 * inv_l;
  yr[j + 16] = o1 * inv_l;
  yr[j + 32] = o2 * inv_l;
  yr[j + 48] = o3 * inv_l;
}

// ---------------------------------------------------------------------------
extern "C" void kernel_launch(void* const* d_in, const int* in_sizes, int n_in,
                              void* d_out, int out_size, void* d_ws, size_t ws_size,
                              hipStream_t stream) {
  (void)in_sizes; (void)n_in; (void)out_size; (void)ws_size;
  const float* x        = (const float*)d_in[0];
  const int*   bias_mat = (const int*)  d_in[1];
  const float* ln1_w    = (const float*)d_in[2];
  const float* ln1_b    = (const float*)d_in[3];
  const float* w_attn_w = (const float*)d_in[4];
  const float* w_attn_b = (const float*)d_in[5];
  const float* w_proj_w = (const float*)d_in[6];
  const float* w_proj_b = (const float*)d_in[7];
  const float* ab_emb   = (const float*)d_in[8];
  const float* edge_emb = (const float*)d_in[9];
  const float* w_ek_w   = (const float*)d_in[10];
  const float* w_ek_b   = (const float*)d_in[11];
  const float* w_ev_w   = (const float*)d_in[12];
  const float* w_ev_b   = (const float*)d_in[13];
  const float* ln2_w    = (const float*)d_in[14];
  const float* ln2_b    = (const float*)d_in[15];
  const float* c_fc_w   = (const float*)d_in[16];
  const float* c_fc_b   = (const float*)d_in[17];
  const float* c_proj_w = (const float*)d_in[18];
  const float* c_proj_b = (const float*)d_in[19];
  float* out = (float*)d_out;

  float* ws = (float*)d_ws;
  float* h1     = ws;                          // 384*512
  float* qkv    = h1     + T_DIM * C_DIM;      // 384*1536
  float* edge_k = qkv    + T_DIM * 3 * C_DIM;  // 16*512
  float* edge_v = edge_k + E_DIM * C_DIM;      // 16*512
  float* yatt   = edge_v + E_DIM * C_DIM;      // 384*512
  float* x1     = yatt   + T_DIM * C_DIM;      // 384*512
  float* h2     = x1     + T_DIM * C_DIM;      // 384*512
  float* hfc    = h2     + T_DIM * C_DIM;      // 384*2048

  // 1) LN1
  ln_kernel<<<dim3(T_DIM), dim3(256), 0, stream>>>(x, ln1_w, ln1_b, h1);
  // 2) QKV = h1 @ w_attn_w.T + b        (384 x 1536 x 512)
  gemm_kernel<0,0><<<dim3(1536/128, (T_DIM+63)/64), dim3(256), 0, stream>>>(
      h1, w_attn_w, w_attn_b, nullptr, qkv, T_DIM, 3*C_DIM, C_DIM);
  // 3) edge_k = edge_emb @ w_edge_k_w.T + b   (16 x 512 x 512), same for v
  gemm_kernel<0,0><<<dim3(C_DIM/128, 1), dim3(256), 0, stream>>>(
      edge_emb, w_ek_w, w_ek_b, nullptr, edge_k, E_DIM, C_DIM, C_DIM);
  gemm_kernel<0,0><<<dim3(C_DIM/128, 1), dim3(256), 0, stream>>>(
      edge_emb, w_ev_w, w_ev_b, nullptr, edge_v, E_DIM, C_DIM, C_DIM);
  // 4) edge-modulated causal attention -> yatt
  attn_kernel<<<dim3(T_DIM/16, H_DIM), dim3(256), 0, stream>>>(
      qkv, bias_mat, edge_k, edge_v, ab_emb, yatt);
  // 5) x1 = x + yatt @ w_proj_w.T + b   (384 x 512 x 512)
  gemm_kernel<0,1><<<dim3(C_DIM/128, (T_DIM+63)/64), dim3(256), 0, stream>>>(
      yatt, w_proj_w, w_proj_b, x, x1, T_DIM, C_DIM, C_DIM);
  // 6) LN2
  ln_kernel<<<dim3(T_DIM), dim3(256), 0, stream>>>(x1, ln2_w, ln2_b, h2);
  // 7) hfc = gelu(h2 @ c_fc_w.T + b)    (384 x 2048 x 512)
  gemm_kernel<1,0><<<dim3(2048/128, (T_DIM+63)/64), dim3(256), 0, stream>>>(
      h2, c_fc_w, c_fc_b, nullptr, hfc, T_DIM, 4*C_DIM, C_DIM);
  // 8) out = x1 + hfc @ c_proj_w.T + b  (384 x 512 x 2048)
  gemm_kernel<0,1><<<dim3(C_DIM/128, (T_DIM+63)/64), dim3(256), 0, stream>>>(
      hfc, c_proj_w, c_proj_b, x1, out, T_DIM, C_DIM, 4*C_DIM);
}